// VertexAlign_22986664968361
// MI455X (gfx1250) — compile-verified
//
#include <hip/hip_runtime.h>
#include <stdint.h>

typedef float v4f __attribute__((ext_vector_type(4)));
typedef unsigned int u32x4 __attribute__((ext_vector_type(4)));
typedef int i32x4 __attribute__((ext_vector_type(4)));
typedef int i32x8 __attribute__((ext_vector_type(8)));

#define WAVE 32

// ---------------------------------------------------------------------------
// Transpose [C, P] -> [P, C] per batch image. The 32x32 tile is brought into
// LDS by a single Tensor Data Mover DMA (tensor_load_to_lds, TENSORcnt),
// using the D# pad feature (1 DWORD pad per 32 DWORDs) to get a 33-float row
// pitch (bank-conflict-free transposed reads). Edge tiles clamp tile_dim and
// disable padding (pitch = tile_dim0).
// ---------------------------------------------------------------------------
__global__ void __launch_bounds__(256)
transpose_cp(const float* __restrict__ in, float* __restrict__ outp, int C, int P) {
  __shared__ float tile[32 * 33];
  const int b = blockIdx.z;
  const int p0 = blockIdx.x * 32;  // dim0 (row length P)
  const int c0 = blockIdx.y * 32;  // dim1 (rows C)
  const float* __restrict__ src = in + (size_t)b * (size_t)C * (size_t)P;
  float* __restrict__ dst = outp + (size_t)b * (size_t)C * (size_t)P;
  const int tx = threadIdx.x;  // 0..31
  const int ty = threadIdx.y;  // 0..7 (wave id)
  const int td0 = min(32, P - p0);  // tile extent along p
  const int td1 = min(32, C - c0);  // tile extent along c
  const bool full = (td0 == 32);
  const int pitch = full ? 33 : td0;

  if (ty == 0) {  // TDM ignores EXEC: issue from wave 0 only
    const uint64_t ga = (uint64_t)(uintptr_t)(src + (size_t)c0 * P + p0);
    const unsigned lds = (unsigned)(uintptr_t)(&tile[0]);

    u32x4 g0;
    g0[0] = 1u;                                   // count=1 (valid), no gather
    g0[1] = lds;                                  // lds_addr (bytes)
    g0[2] = (unsigned)(ga & 0xffffffffu);         // global_addr[31:0]
    g0[3] = (unsigned)((ga >> 32) & 0x01ffffffu)  // global_addr[56:32]
            | (2u << 30);                         // type=2 ("image")

    unsigned flags = (2u << 16);                  // data_size = 4 bytes
    if (full) flags |= (1u << 20)                 // pad_enable
                     | (4u << 22);                // pad_interval: 32 DWORDs
                                                  // pad_amount=0 -> 1 DWORD
    i32x8 g1;
    g1[0] = (int)flags;                                        // mask=0, flags
    g1[1] = (int)(((unsigned)P & 0xffffu) << 16);              // tdim0[15:0]
    g1[2] = (int)((((unsigned)P >> 16) & 0xffffu) |
                  (((unsigned)C & 0xffffu) << 16));            // tdim0 hi|tdim1 lo
    g1[3] = (int)((((unsigned)C >> 16) & 0xffffu) |
                  (((unsigned)td0 & 0xffffu) << 16));          // tdim1 hi|tile_dim0
    g1[4] = (int)((unsigned)td1 & 0xffffu);                    // tile_dim1, tile_dim2=0
    g1[5] = (int)P;                                            // tdim0_stride[31:0]
    g1[6] = 0;                                                 // stride hi / dim1_stride lo
    g1[7] = 0;

    const i32x4 gz = (i32x4)0;  // groups 2/3 unused (2-D tensor)
#if __clang_major__ >= 23
    const i32x8 gz8 = (i32x8)0;
    __builtin_amdgcn_tensor_load_to_lds(g0, g1, gz, gz, gz8, 0);
#else
    __builtin_amdgcn_tensor_load_to_lds(g0, g1, gz, gz, 0);
#endif
    __builtin_amdgcn_s_wait_tensorcnt(0);
  }
  __syncthreads();

  // Coalesced transposed stores: dst[(p0+i)*C + (c0+tx)] = in[c0+tx][p0+i].
#pragma unroll
  for (int i0 = 0; i0 < 32; i0 += 8) {
    const int i = i0 + ty;
    if (i < td0 && tx < td1)
      dst[(size_t)(p0 + i) * C + (c0 + tx)] = tile[tx * pitch + i];
  }
}

// ---------------------------------------------------------------------------
// Per-scale index math (replicates the reference's degenerate {0,1} weights).
// ---------------------------------------------------------------------------
template <int S>
__device__ __forceinline__ bool scale_idx(float h, float w, int& off) {
  constexpr float sc = (float)S / 224.0f;  // exact power-of-two reciprocal
  const float x = h * sc;
  const float y = w * sc;
  const int x1 = (int)floorf(x);
  const int y1 = (int)floorf(y);
  const int x2 = min((int)ceilf(x), S - 1);
  const int y2 = min((int)ceilf(y), S - 1);
  off = x1 * S + y1;
  return (x2 > x1) && (y2 > y1);
}

// Segment copy from a transposed map t[b][P][C]: contiguous row -> contiguous out.
template <int C, int S>
__device__ __forceinline__ v4f* copy_seg_tr(const float* __restrict__ t, int b,
                                            float h, float w, int lane,
                                            v4f* __restrict__ dst) {
  int off;
  const bool valid = scale_idx<S>(h, w, off);
  const v4f* __restrict__ src =
      (const v4f*)(t + ((size_t)b * (S * S) + (size_t)off) * C);
  __builtin_prefetch((const void*)(src + lane), 0, 0);  // global_prefetch
  const v4f z = {0.0f, 0.0f, 0.0f, 0.0f};
#pragma unroll
  for (int j = lane; j < C / 4; j += WAVE) {
    v4f r = z;
    if (valid) r = src[j];
    __builtin_nontemporal_store(r, dst + j);  // write-once 1GB: keep L2 for fmaps
  }
  return dst + C / 4;
}

// Fallback segment copy straight from [C,S,S] layout (strided channel gather).
template <int C, int S>
__device__ __forceinline__ v4f* copy_seg_direct(const float* __restrict__ f, int b,
                                                float h, float w, int lane,
                                                v4f* __restrict__ dst) {
  int off;
  const bool valid = scale_idx<S>(h, w, off);
  const float* __restrict__ src = f + (size_t)b * C * (S * S) + off;
  const v4f z = {0.0f, 0.0f, 0.0f, 0.0f};
#pragma unroll
  for (int j = lane; j < C / 4; j += WAVE) {
    v4f r = z;
    if (valid) {
      const int c = 4 * j;
      r.x = src[(size_t)(c + 0) * (S * S)];
      r.y = src[(size_t)(c + 1) * (S * S)];
      r.z = src[(size_t)(c + 2) * (S * S)];
      r.w = src[(size_t)(c + 3) * (S * S)];
    }
    __builtin_nontemporal_store(r, dst + j);
  }
  return dst + C / 4;
}

// ---------------------------------------------------------------------------
// Hot kernel: one wave32 per vertex. All control state (b, off, valid) is
// wave-uniform; 30 fully-coalesced b128 loads + b128 NT stores per lane.
// ---------------------------------------------------------------------------
__global__ void __launch_bounds__(256)
vertex_align_gather_tr(const float* __restrict__ pos,
                       const float* __restrict__ t0, const float* __restrict__ t1,
                       const float* __restrict__ t2, const float* __restrict__ t3,
                       float* __restrict__ out, int BV, int V) {
  const int wave = blockIdx.x * (blockDim.x >> 5) + (threadIdx.x >> 5);
  const int lane = threadIdx.x & (WAVE - 1);
  if (wave >= BV) return;
  const float* __restrict__ p = pos + 3 * (size_t)wave;
  const float px = p[0], py = p[1], pz = p[2];
  const float h = fminf(fmaxf(248.0f * (py / pz) + 111.5f, 0.0f), 223.0f);
  const float w = fminf(fmaxf(248.0f * (px / (-pz)) + 111.5f, 0.0f), 223.0f);
  const int b = wave / V;
  v4f* dst = (v4f*)(out + (size_t)wave * 3840);
  dst = copy_seg_tr<256, 56>(t0, b, h, w, lane, dst);
  dst = copy_seg_tr<512, 28>(t1, b, h, w, lane, dst);
  dst = copy_seg_tr<1024, 14>(t2, b, h, w, lane, dst);
  (void)copy_seg_tr<2048, 7>(t3, b, h, w, lane, dst);
}

__global__ void __launch_bounds__(256)
vertex_align_gather_direct(const float* __restrict__ pos,
                           const float* __restrict__ f0, const float* __restrict__ f1,
                           const float* __restrict__ f2, const float* __restrict__ f3,
                           float* __restrict__ out, int BV, int V) {
  const int wave = blockIdx.x * (blockDim.x >> 5) + (threadIdx.x >> 5);
  const int lane = threadIdx.x & (WAVE - 1);
  if (wave >= BV) return;
  const float* __restrict__ p = pos + 3 * (size_t)wave;
  const float px = p[0], py = p[1], pz = p[2];
  const float h = fminf(fmaxf(248.0f * (py / pz) + 111.5f, 0.0f), 223.0f);
  const float w = fminf(fmaxf(248.0f * (px / (-pz)) + 111.5f, 0.0f), 223.0f);
  const int b = wave / V;
  v4f* dst = (v4f*)(out + (size_t)wave * 3840);
  dst = copy_seg_direct<256, 56>(f0, b, h, w, lane, dst);
  dst = copy_seg_direct<512, 28>(f1, b, h, w, lane, dst);
  dst = copy_seg_direct<1024, 14>(f2, b, h, w, lane, dst);
  (void)copy_seg_direct<2048, 7>(f3, b, h, w, lane, dst);
}

// ---------------------------------------------------------------------------
extern "C" void kernel_launch(void* const* d_in, const int* in_sizes, int n_in,
                              void* d_out, int out_size, void* d_ws, size_t ws_size,
                              hipStream_t stream) {
  const float* f0 = (const float*)d_in[0];
  const float* f1 = (const float*)d_in[1];
  const float* f2 = (const float*)d_in[2];
  const float* f3 = (const float*)d_in[3];
  const float* pos = (const float*)d_in[4];
  // in_sizes[5] (verts_per_mesh) lives on device; derive shapes from sizes.
  int B = in_sizes[0] / (256 * 56 * 56);
  if (B <= 0) B = 1;
  const int BV = in_sizes[4] / 3;  // total vertices across batch
  const int V = BV / B;
  if (BV <= 0) return;
  float* out = (float*)d_out;

  const size_t n0 = (size_t)B * 256 * 3136;
  const size_t n1 = (size_t)B * 512 * 784;
  const size_t n2 = (size_t)B * 1024 * 196;
  const size_t n3 = (size_t)B * 2048 * 49;
  const size_t need = (n0 + n1 + n2 + n3) * sizeof(float);

  const int wavesPerBlock = 256 / WAVE;  // 8 vertices per block
  const int blocks = (BV + wavesPerBlock - 1) / wavesPerBlock;

  if (ws_size >= need) {
    float* t0 = (float*)d_ws;
    float* t1 = t0 + n0;
    float* t2 = t1 + n1;
    float* t3 = t2 + n2;
    const dim3 blk(32, 8, 1);
    transpose_cp<<<dim3((3136 + 31) / 32, (256 + 31) / 32, B), blk, 0, stream>>>(f0, t0, 256, 3136);
    transpose_cp<<<dim3((784 + 31) / 32, (512 + 31) / 32, B), blk, 0, stream>>>(f1, t1, 512, 784);
    transpose_cp<<<dim3((196 + 31) / 32, (1024 + 31) / 32, B), blk, 0, stream>>>(f2, t2, 1024, 196);
    transpose_cp<<<dim3((49 + 31) / 32, (2048 + 31) / 32, B), blk, 0, stream>>>(f3, t3, 2048, 49);
    vertex_align_gather_tr<<<blocks, 256, 0, stream>>>(pos, t0, t1, t2, t3, out, BV, V);
  } else {
    vertex_align_gather_direct<<<blocks, 256, 0, stream>>>(pos, f0, f1, f2, f3, out, BV, V);
  }
}